// OVCondNMSPostProcess_13408887899034
// MI455X (gfx1250) — compile-verified
//
#include <hip/hip_runtime.h>
#include <hip/hip_bf16.h>

typedef __attribute__((ext_vector_type(2))) float v2f;
typedef __attribute__((ext_vector_type(8))) float v8f;

#define NQ        900
#define TOPK_PRE  100
#define TOPK_POST 20
#define SORT_N    1024
#define IOU_THR   0.7f

#if __has_builtin(__builtin_amdgcn_global_load_async_to_lds_b32)
#define HAVE_ASYNC_LDS 1
#else
#define HAVE_ASYNC_LDS 0
#endif

#if __has_builtin(__builtin_amdgcn_wmma_f32_16x16x4_f32)
#define HAVE_WMMA_F32X4 1
#else
#define HAVE_WMMA_F32X4 0
#endif

typedef __attribute__((address_space(1))) int* gptr_i32;
typedef __attribute__((address_space(3))) int* lptr_i32;

// One workgroup per (b,p) problem: 1024 blocks x 256 threads (8 wave32 / block).
__global__ __launch_bounds__(256) void nms_post_kernel(
    const float* __restrict__ pred_logits,   // (16,1,57600,2)
    const float* __restrict__ pred_boxes,    // (16,1,57600,4)
    const float* __restrict__ target_sizes,  // (16,2) = (h,w)
    const int*   __restrict__ labels,        // (16,64)
    float* __restrict__ out)                 // sel_s|sel_b|sel_l|vmask flat
{
  const int tid = threadIdx.x;
  const int bp  = blockIdx.x;          // 0..1023
  const int b   = bp >> 6;
  const int p   = bp & 63;

  const float* lg = pred_logits + ((size_t)b * 57600 + (size_t)p * 900) * 2;
  const float* bx = pred_boxes  + ((size_t)b * 57600 + (size_t)p * 900) * 4;
  const float img_h = target_sizes[b * 2 + 0];
  const float img_w = target_sizes[b * 2 + 1];
  const int   lab   = labels[b * 64 + p];

  __shared__ float    skey[SORT_N];        // scores (sort keys)
  __shared__ int      sidx[SORT_N];        // query indices
  __shared__ float    sbox[112][4];        // top-100 cxcywh (padded to 112)
  __shared__ float    sxyxy[112][4];       // scaled xyxy
  __shared__ unsigned ssup[TOPK_PRE * 4];  // suppress bitmask rows
  __shared__ unsigned skeep[4];            // final keep bitmask

  // Prefetch the 14.4KB box slab for this problem toward L2 while sorting.
  if (tid < (NQ * 16 + 63) / 64)
    __builtin_prefetch((const char*)bx + tid * 64, 0, 1);

  // ---- Stage 1: logits (channel 1) -> LDS, then sigmoid -------------------
#if HAVE_ASYNC_LDS
  for (int q = tid; q < NQ; q += 256) {
    __builtin_amdgcn_global_load_async_to_lds_b32(
        (gptr_i32)(lg + 2 * q + 1),
        (lptr_i32)&skey[q], 0, 0);
  }
#if __has_builtin(__builtin_amdgcn_s_wait_asynccnt)
  __builtin_amdgcn_s_wait_asynccnt(0);
#else
  asm volatile("s_wait_asynccnt 0" ::: "memory");
#endif
  __syncthreads();
  for (int q = tid; q < SORT_N; q += 256) {
    float s = -1e30f;
    if (q < NQ) { float x = skey[q]; s = 1.0f / (1.0f + __expf(-x)); }
    skey[q] = s;
    sidx[q] = q;
  }
#else
  for (int q = tid; q < SORT_N; q += 256) {
    float s = -1e30f;
    if (q < NQ) { float x = lg[2 * q + 1]; s = 1.0f / (1.0f + __expf(-x)); }
    skey[q] = s;
    sidx[q] = q;
  }
#endif
  __syncthreads();

  // ---- Stage 2: bitonic sort 1024 descending (2 CAS / thread / stage) -----
  for (int k = 2; k <= SORT_N; k <<= 1) {
    for (int j = k >> 1; j > 0; j >>= 1) {
      for (int i = tid; i < SORT_N; i += 256) {
        int ixj = i ^ j;
        if (ixj > i) {
          float a = skey[i], c = skey[ixj];
          bool desc = ((i & k) == 0);
          bool sw = desc ? (a < c) : (a > c);
          if (sw) {
            skey[i] = c; skey[ixj] = a;
            int t = sidx[i]; sidx[i] = sidx[ixj]; sidx[ixj] = t;
          }
        }
      }
      __syncthreads();
    }
  }

  // ---- Stage 3: gather top-100 boxes (b128 loads), pad to 112 -------------
  if (tid < 112) {
    float cx = 0.f, cy = 0.f, w = 0.f, h = 0.f;
    if (tid < TOPK_PRE) {
      int q = sidx[tid];
      const float4 v = *(const float4*)(bx + 4 * q);  // 16B-aligned
      cx = v.x; cy = v.y; w = v.z; h = v.w;
    }
    sbox[tid][0] = cx; sbox[tid][1] = cy; sbox[tid][2] = w; sbox[tid][3] = h;
  }
  __syncthreads();

  // ---- Stage 4: cxcywh -> xyxy * (w,h,w,h) as (16x4)x(4x16) WMMA ----------
  // xyxy[n][j] = sum_k box[n][k] * B[k][j];
  //   B = [[ W, 0,  W, 0],
  //        [ 0, H,  0, H],
  //        [-W/2,0, W/2,0],
  //        [ 0,-H/2,0, H/2]] (cols 4..15 zero)
#if HAVE_WMMA_F32X4
  if (tid < 32) {                       // wave 0, EXEC all-ones
    const int  m  = tid & 15;
    const bool hi = tid >= 16;
    // A layout 16x4 f32: v0={K0 lanes0-15, K2 lanes16-31}, v1={K1, K3}
    // B layout 4x16 f32: v0={row0 lanes0-15, row2 lanes16-31}, v1={row1, row3}
    v2f B;
    if (!hi) {
      B.x = (m == 0 || m == 2) ? img_w : 0.f;                  // row 0
      B.y = (m == 1 || m == 3) ? img_h : 0.f;                  // row 1
    } else {
      B.x = (m == 0) ? -0.5f * img_w : ((m == 2) ? 0.5f * img_w : 0.f); // row 2
      B.y = (m == 1) ? -0.5f * img_h : ((m == 3) ? 0.5f * img_h : 0.f); // row 3
    }
    for (int t = 0; t < 7; ++t) {       // 7 tiles of 16 boxes = 112
      int row = t * 16 + m;
      v2f A;
      A.x = hi ? sbox[row][2] : sbox[row][0];  // K2=w / K0=cx
      A.y = hi ? sbox[row][3] : sbox[row][1];  // K3=h / K1=cy
      v8f C = {};
      v8f D = __builtin_amdgcn_wmma_f32_16x16x4_f32(
          false, A, false, B, (short)0, C, false, false);
      // D: vgpr r: lanes0-15 -> (M=r, N=lane); lanes16-31 -> (M=8+r, N=lane-16)
      if (m < 4) {
        int rb = t * 16 + (hi ? 8 : 0);
#pragma unroll
        for (int r = 0; r < 8; ++r) sxyxy[rb + r][m] = D[r];
      }
    }
  }
#else
  if (tid < 112) {
    float cx = sbox[tid][0], cy = sbox[tid][1], w = sbox[tid][2], h = sbox[tid][3];
    sxyxy[tid][0] = (cx - 0.5f * w) * img_w;
    sxyxy[tid][1] = (cy - 0.5f * h) * img_h;
    sxyxy[tid][2] = (cx + 0.5f * w) * img_w;
    sxyxy[tid][3] = (cy + 0.5f * h) * img_h;
  }
#endif
  __syncthreads();

  // ---- Stage 5: 100x100 IoU suppress bitmasks (400 tasks x 32 IoUs) -------
  for (int task = tid; task < TOPK_PRE * 4; task += 256) {
    int i = task >> 2, w = task & 3;
    float ax1 = sxyxy[i][0], ay1 = sxyxy[i][1], ax2 = sxyxy[i][2], ay2 = sxyxy[i][3];
    float aarea = fmaxf(ax2 - ax1, 0.f) * fmaxf(ay2 - ay1, 0.f);
    unsigned bits = 0u;
    int j0 = w * 32;
#pragma unroll 4
    for (int bb = 0; bb < 32; ++bb) {
      int j = j0 + bb;
      if (j > i && j < TOPK_PRE) {
        float bx1 = sxyxy[j][0], by1 = sxyxy[j][1], bx2 = sxyxy[j][2], by2 = sxyxy[j][3];
        float iw = fmaxf(fminf(ax2, bx2) - fmaxf(ax1, bx1), 0.f);
        float ih = fmaxf(fminf(ay2, by2) - fmaxf(ay1, by1), 0.f);
        float inter = iw * ih;
        float barea = fmaxf(bx2 - bx1, 0.f) * fmaxf(by2 - by1, 0.f);
        float iou = inter / (aarea + barea - inter + 1e-7f);
        if (iou > IOU_THR) bits |= (1u << bb);
      }
    }
    ssup[task] = bits;
  }
  __syncthreads();

  // ---- Stage 6: serial greedy keep (bitmask form, 4x u32 in registers) ----
  if (tid == 0) {
    unsigned kp0 = ~0u, kp1 = ~0u, kp2 = ~0u, kp3 = 0xFu;  // bits 0..99
    for (int i = 0; i < TOPK_PRE; ++i) {
      unsigned kw = (i < 32) ? kp0 : (i < 64) ? kp1 : (i < 96) ? kp2 : kp3;
      if ((kw >> (i & 31)) & 1u) {
        kp0 &= ~ssup[i * 4 + 0];
        kp1 &= ~ssup[i * 4 + 1];
        kp2 &= ~ssup[i * 4 + 2];
        kp3 &= ~ssup[i * 4 + 3];
      }
    }
    skeep[0] = kp0; skeep[1] = kp1; skeep[2] = kp2; skeep[3] = kp3;
  }
  __syncthreads();

  // ---- Stage 7: rank-compact to 20 slots and write outputs ----------------
  const size_t base_s = (size_t)bp * TOPK_POST;
  const size_t base_b = 20480 + (size_t)bp * TOPK_POST * 4;
  const size_t base_l = 102400 + (size_t)bp * TOPK_POST;
  const size_t base_m = 122880 + (size_t)bp * TOPK_POST;

  if (tid < TOPK_POST) {  // zero-init (d_out is poisoned by harness)
    out[base_s + tid] = 0.f;
    out[base_l + tid] = -1.f;
    out[base_m + tid] = 0.f;
#pragma unroll
    for (int c = 0; c < 4; ++c) out[base_b + tid * 4 + c] = 0.f;
  }
  __syncthreads();

  if (tid < TOPK_PRE) {
    int wsel = tid >> 5, bbit = tid & 31;
    unsigned kw = skeep[wsel];
    bool kept = (kw >> bbit) & 1u;
    int rank = 0;
    for (int w = 0; w < wsel; ++w) rank += __popc(skeep[w]);
    rank += __popc(kw & ((1u << bbit) - 1u));
    if (kept && rank < TOPK_POST) {
      out[base_s + rank] = skey[tid];
      out[base_l + rank] = (float)lab;
      out[base_m + rank] = 1.f;
#pragma unroll
      for (int c = 0; c < 4; ++c) out[base_b + rank * 4 + c] = sxyxy[tid][c];
    }
  }
}

extern "C" void kernel_launch(void* const* d_in, const int* in_sizes, int n_in,
                              void* d_out, int out_size, void* d_ws, size_t ws_size,
                              hipStream_t stream) {
  const float* pred_logits  = (const float*)d_in[0];
  const float* pred_boxes   = (const float*)d_in[1];
  const float* target_sizes = (const float*)d_in[2];
  const int*   labels       = (const int*)d_in[3];
  float* out = (float*)d_out;
  (void)in_sizes; (void)n_in; (void)out_size; (void)d_ws; (void)ws_size;

  nms_post_kernel<<<dim3(1024), dim3(256), 0, stream>>>(
      pred_logits, pred_boxes, target_sizes, labels, out);
}